// A3TGCN_2_62569083568442
// MI455X (gfx1250) — compile-verified
//
#include <hip/hip_runtime.h>
#include <cstdint>

// Problem constants (fixed by the reference)
#define Bsz 16
#define Nn  5000
#define Ff  8
#define Tt  12
#define Ee  160000
#define OC  64
#define NT  ((Nn + 15) / 16)   // 313 node tiles

typedef float v2f __attribute__((ext_vector_type(2)));
typedef float v8f __attribute__((ext_vector_type(8)));

// ---------------------------------------------------------------- CSR build
__global__ void k_init(int* __restrict__ cnt, int* __restrict__ cursor) {
  int i = blockIdx.x * blockDim.x + threadIdx.x;
  if (i < Nn) { cnt[i] = 0; cursor[i] = 0; }
}

__global__ void k_count(const int* __restrict__ ei, int* __restrict__ cnt) {
  int e = blockIdx.x * blockDim.x + threadIdx.x;
  if (e < Ee) atomicAdd(&cnt[ei[Ee + e]], 1);   // dst row
}

// Single-block scan: rowptr (exclusive prefix of in-degree) + dinv = rsqrt(deg+1 self loop)
__global__ __launch_bounds__(1024) void k_scan(const int* __restrict__ cnt,
                                               int* __restrict__ rowptr,
                                               float* __restrict__ dinv) {
  __shared__ int sums[1024];
  const int t = threadIdx.x;
  const int base = t * 5;           // 1024*5 = 5120 >= 5000
  int loc[5];
  int tot = 0;
#pragma unroll
  for (int k = 0; k < 5; ++k) {
    int idx = base + k;
    int v = (idx < Nn) ? cnt[idx] : 0;
    loc[k] = tot; tot += v;
  }
  sums[t] = tot;
  __syncthreads();
  for (int off = 1; off < 1024; off <<= 1) {
    int v = (t >= off) ? sums[t - off] : 0;
    __syncthreads();
    sums[t] += v;
    __syncthreads();
  }
  int basesum = (t == 0) ? 0 : sums[t - 1];
#pragma unroll
  for (int k = 0; k < 5; ++k) {
    int idx = base + k;
    if (idx < Nn) rowptr[idx] = basesum + loc[k];
  }
  if (t == 1023) rowptr[Nn] = sums[1023];
  for (int idx = t; idx < Nn; idx += 1024)
    dinv[idx] = rsqrtf((float)(cnt[idx] + 1));   // +1 = self loop
}

__global__ void k_fill(const int* __restrict__ ei, const int* __restrict__ rowptr,
                       int* __restrict__ cursor, int* __restrict__ eid) {
  int e = blockIdx.x * blockDim.x + threadIdx.x;
  if (e < Ee) {
    int d = ei[Ee + e];
    int pos = rowptr[d] + atomicAdd(&cursor[d], 1);
    eid[pos] = e;
  }
}

// Sort each node's edge list by edge id (bitwise-deterministic sum order),
// then materialize src index + symmetric GCN norm.
__global__ void k_sortnorm(const int* __restrict__ ei, const int* __restrict__ rowptr,
                           const float* __restrict__ dinv, int* __restrict__ eid,
                           int* __restrict__ csrc, float* __restrict__ cnorm) {
  int n = blockIdx.x * blockDim.x + threadIdx.x;
  if (n >= Nn) return;
  int e0 = rowptr[n], e1 = rowptr[n + 1];
  for (int i = e0 + 1; i < e1; ++i) {          // tiny segments (~32 avg)
    int key = eid[i]; int j = i - 1;
    while (j >= e0 && eid[j] > key) { eid[j + 1] = eid[j]; --j; }
    eid[j + 1] = key;
  }
  float dn = dinv[n];
  for (int i = e0; i < e1; ++i) {
    int s = ei[eid[i]];
    csrc[i] = s;
    cnorm[i] = dinv[s] * dn;
  }
}

// ------------------------------------------------- fold weights + softmax(att)
__global__ void k_fuse(const float* __restrict__ Wz, const float* __restrict__ bz,
                       const float* __restrict__ Lz, const float* __restrict__ lbz,
                       const float* __restrict__ Wh, const float* __restrict__ bh,
                       const float* __restrict__ Lh, const float* __restrict__ lbh,
                       const float* __restrict__ att, const float* __restrict__ fcW,
                       float* __restrict__ Wzp, float* __restrict__ bzp,
                       float* __restrict__ Whp, float* __restrict__ bhp,
                       float* __restrict__ attw, float* __restrict__ fcWp) {
  int c = threadIdx.x;  // 0..63
  for (int f = 0; f < Ff; ++f) {
    float sz = 0.f, sh = 0.f;
    for (int k = 0; k < OC; ++k) {
      sz += Wz[f * OC + k] * Lz[k * OC + c];   // only first OC rows of L matter (H==0)
      sh += Wh[f * OC + k] * Lh[k * OC + c];
    }
    Wzp[f * OC + c] = sz;
    Whp[f * OC + c] = sh;
  }
  float sz = lbz[c], sh = lbh[c];
  for (int k = 0; k < OC; ++k) {
    sz += bz[k] * Lz[k * OC + c];
    sh += bh[k] * Lh[k * OC + c];
  }
  bzp[c] = sz; bhp[c] = sh;
  if (c < 16)
    for (int k = 0; k < OC; ++k)
      fcWp[k * 16 + c] = (c < Tt) ? fcW[k * Tt + c] : 0.f;   // pad N 12 -> 16
  if (c == 0) {
    float m = att[0];
    for (int t = 1; t < Tt; ++t) m = fmaxf(m, att[t]);
    float s = 0.f, e[Tt];
    for (int t = 0; t < Tt; ++t) { e[t] = __expf(att[t] - m); s += e[t]; }
    for (int t = 0; t < Tt; ++t) attw[t] = e[t] / s;
  }
}

// ---------------------------------------------- feature-space GCN aggregation
// agg[b][n][t][f] = dinv[n]^2 * x[b,n,f,t] + sum_e norm_e * x[b,src_e,f,t]
// One 96-thread block per (n,b); gathers are L2-resident (1.9MB x-slice per b).
__global__ __launch_bounds__(96) void k_agg(const float* __restrict__ x,
                                            const int* __restrict__ rowptr,
                                            const int* __restrict__ csrc,
                                            const float* __restrict__ cnorm,
                                            const float* __restrict__ dinv,
                                            float* __restrict__ agg) {
  const int ch = threadIdx.x;          // ch = f*12 + t  (x channel order)
  const int n = blockIdx.x, b = blockIdx.y;
  const float dn = dinv[n];
  const size_t bbase = (size_t)b * Nn * (Ff * Tt);
  float acc = dn * dn * x[bbase + (size_t)n * (Ff * Tt) + ch];
  const int e1 = rowptr[n + 1];
  for (int e = rowptr[n]; e < e1; ++e)
    acc += cnorm[e] * x[bbase + (size_t)csrc[e] * (Ff * Tt) + ch];
  const int f = ch / Tt, t = ch % Tt;  // transpose to [t][f] for WMMA A loads
  agg[((size_t)(b * Nn + n) * Tt + t) * Ff + f] = acc;
}

// ------------------------------------------------------- fused WMMA epilogue
// One wave per 16-node tile. Per t: Az/Ah = G_t(16x8) @ W'(8x64) via
// v_wmma_f32_16x16x4_f32 (2 k-steps x 4 col-tiles x 2 gates), branch-free
// activation + attention accumulate; then out = Hacc(16x64) @ fcWpad(64x16).
// A-row M only feeds C/D-row M, so OOB node rows (clamped pointer) produce
// garbage that is never stored -- no per-lane guards, EXEC stays all-1s.
__global__ __launch_bounds__(32) void k_main(const float* __restrict__ agg,
                                             const float* __restrict__ Wzp,
                                             const float* __restrict__ bzp,
                                             const float* __restrict__ Whp,
                                             const float* __restrict__ bhp,
                                             const float* __restrict__ attw,
                                             const float* __restrict__ fcWp,
                                             const float* __restrict__ fcb,
                                             float* __restrict__ out) {
  __shared__ float lds[16 * 68];       // 16x64 Hacc tile, pitch 68 vs bank conflicts
  const int lane = threadIdx.x;
  const int half = lane >> 4;          // 0: K pair {0,1}; 1: K pair {2,3}
  const int lr = lane & 15;
  const int n0 = blockIdx.x * 16;
  const int b = blockIdx.y;

  // B operands: B(4x16) layout = lane<16 rows{kb,kb+1}, lane>=16 rows{kb+2,kb+3}, col=lr
  v2f Bz[2][4], Bh[2][4];
#pragma unroll
  for (int kb2 = 0; kb2 < 2; ++kb2) {
    const int kr = kb2 * 4 + 2 * half;
#pragma unroll
    for (int ct = 0; ct < 4; ++ct) {
      const int col = ct * 16 + lr;
      Bz[kb2][ct].x = Wzp[kr * OC + col];
      Bz[kb2][ct].y = Wzp[(kr + 1) * OC + col];
      Bh[kb2][ct].x = Whp[kr * OC + col];
      Bh[kb2][ct].y = Whp[(kr + 1) * OC + col];
    }
  }
  float biasz[4], biash[4];
#pragma unroll
  for (int ct = 0; ct < 4; ++ct) {
    biasz[ct] = bzp[ct * 16 + lr];
    biash[ct] = bhp[ct * 16 + lr];
  }

  v8f z8;
#pragma unroll
  for (int j = 0; j < 8; ++j) z8[j] = 0.f;
  v8f Hacc[4];
#pragma unroll
  for (int ct = 0; ct < 4; ++ct) Hacc[ct] = z8;

  const int node = n0 + lr;
  const int nodec = node < Nn ? node : (Nn - 1);        // clamp: garbage rows unstored
  const float* gbase = agg + ((size_t)(b * Nn + nodec) * Tt) * Ff;

#pragma unroll
  for (int t = 0; t < Tt; ++t) {
    // A(16x4) layout: lane<16 -> (K0,K1), lane>=16 -> (K2,K3); row M = lr
    const v2f A0 = *(const v2f*)(gbase + t * Ff + 2 * half);      // 8B-aligned b64
    const v2f A1 = *(const v2f*)(gbase + t * Ff + 4 + 2 * half);
    const float at = attw[t];
#pragma unroll
    for (int ct = 0; ct < 4; ++ct) {
      v8f cz = __builtin_amdgcn_wmma_f32_16x16x4_f32(false, A0, false, Bz[0][ct],
                                                     (short)0, z8, false, false);
      cz = __builtin_amdgcn_wmma_f32_16x16x4_f32(false, A1, false, Bz[1][ct],
                                                 (short)0, cz, false, false);
      v8f ch = __builtin_amdgcn_wmma_f32_16x16x4_f32(false, A0, false, Bh[0][ct],
                                                     (short)0, z8, false, false);
      ch = __builtin_amdgcn_wmma_f32_16x16x4_f32(false, A1, false, Bh[1][ct],
                                                 (short)0, ch, false, false);
#pragma unroll
      for (int j = 0; j < 8; ++j) {
        float az = cz[j] + biasz[ct];
        float ah = ch[j] + biash[ct];
        // (1 - sigmoid(az)) = 1/(1+exp(az)); tanh(ah) = 1 - 2/(exp(2*ah)+1).
        // Branch-free: v_exp + v_rcp saturate correctly via inf/0.
        float sig = __builtin_amdgcn_rcpf(1.f + __expf(az));
        float th = 1.f - 2.f * __builtin_amdgcn_rcpf(1.f + __expf(2.f * ah));
        Hacc[ct][j] += at * th * sig;
      }
    }
  }

  // C layout -> LDS row-major: row M = j + 8*half, col = ct*16 + lr
#pragma unroll
  for (int ct = 0; ct < 4; ++ct)
#pragma unroll
    for (int j = 0; j < 8; ++j)
      lds[(j + 8 * half) * 68 + ct * 16 + lr] = Hacc[ct][j];
  __syncthreads();

  // Final fc: D(16x16) = Hacc(16x64) @ fcWpad(64x16), K=4 per step
  v8f D = z8;
#pragma unroll
  for (int kb = 0; kb < OC; kb += 4) {
    const int kr = kb + 2 * half;
    v2f A, Bf;
    A.x = lds[lr * 68 + kr];
    A.y = lds[lr * 68 + kr + 1];
    Bf.x = fcWp[kr * 16 + lr];
    Bf.y = fcWp[(kr + 1) * 16 + lr];
    D = __builtin_amdgcn_wmma_f32_16x16x4_f32(false, A, false, Bf,
                                              (short)0, D, false, false);
  }

  if (lr < Tt) {
    const float fb = fcb[lr];
#pragma unroll
    for (int j = 0; j < 8; ++j) {
      int nr = n0 + j + 8 * half;
      if (nr < Nn) out[(size_t)(b * Nn + nr) * Tt + lr] = D[j] + fb;
    }
  }
}

// --------------------------------------------------------------------- launch
extern "C" void kernel_launch(void* const* d_in, const int* in_sizes, int n_in,
                              void* d_out, int out_size, void* d_ws, size_t ws_size,
                              hipStream_t stream) {
  const float* x   = (const float*)d_in[0];
  const int*   ei  = (const int*)d_in[1];
  const float* Wz  = (const float*)d_in[2];
  const float* bz  = (const float*)d_in[3];
  const float* Lz  = (const float*)d_in[4];
  const float* lbz = (const float*)d_in[5];
  // d_in[6..9] = r-gate params: dead code in the reference (H == 0)
  const float* Wh  = (const float*)d_in[10];
  const float* bh  = (const float*)d_in[11];
  const float* Lh  = (const float*)d_in[12];
  const float* lbh = (const float*)d_in[13];
  const float* att = (const float*)d_in[14];
  const float* fcW = (const float*)d_in[15];
  const float* fcb = (const float*)d_in[16];
  float* out = (float*)d_out;

  char* w = (char*)d_ws;
  size_t o = 0;
  auto take = [&](size_t bytes) -> char* {
    char* p = w + o;
    o += (bytes + 255) & ~(size_t)255;
    return p;
  };
  float* agg    = (float*)take((size_t)Bsz * Nn * Ff * Tt * 4);  // 30.7 MB
  int*   cnt    = (int*)take((size_t)Nn * 4);
  int*   cursor = (int*)take((size_t)Nn * 4);
  int*   rowptr = (int*)take((size_t)(Nn + 1) * 4);
  float* dinv   = (float*)take((size_t)Nn * 4);
  int*   eid    = (int*)take((size_t)Ee * 4);
  int*   csrc   = (int*)take((size_t)Ee * 4);
  float* cnorm  = (float*)take((size_t)Ee * 4);
  float* Wzp    = (float*)take((size_t)Ff * OC * 4);
  float* Whp    = (float*)take((size_t)Ff * OC * 4);
  float* bzp    = (float*)take((size_t)OC * 4);
  float* bhp    = (float*)take((size_t)OC * 4);
  float* attw   = (float*)take((size_t)Tt * 4);
  float* fcWp   = (float*)take((size_t)OC * 16 * 4);

  k_init<<<(Nn + 255) / 256, 256, 0, stream>>>(cnt, cursor);
  k_count<<<(Ee + 255) / 256, 256, 0, stream>>>(ei, cnt);
  k_scan<<<1, 1024, 0, stream>>>(cnt, rowptr, dinv);
  k_fill<<<(Ee + 255) / 256, 256, 0, stream>>>(ei, rowptr, cursor, eid);
  k_sortnorm<<<(Nn + 255) / 256, 256, 0, stream>>>(ei, rowptr, dinv, eid, csrc, cnorm);
  k_fuse<<<1, 64, 0, stream>>>(Wz, bz, Lz, lbz, Wh, bh, Lh, lbh, att, fcW,
                               Wzp, bzp, Whp, bhp, attw, fcWp);
  k_agg<<<dim3(Nn, Bsz), 96, 0, stream>>>(x, rowptr, csrc, cnorm, dinv, agg);
  k_main<<<dim3(NT, Bsz), 32, 0, stream>>>(agg, Wzp, bzp, Whp, bhp, attw, fcWp, fcb, out);
}